// Attention_5815385719367
// MI455X (gfx1250) — compile-verified
//
#include <hip/hip_runtime.h>
#include <hip/hip_bf16.h>

typedef __attribute__((ext_vector_type(2))) float v2f;
typedef __attribute__((ext_vector_type(8))) float v8f;
typedef __attribute__((__vector_size__(4 * sizeof(int)))) int v4i;
typedef __attribute__((address_space(1))) v4i* as1_v4i_p;
typedef __attribute__((address_space(3))) v4i* as3_v4i_p;

#define EMBED   64
#define NM      16384
#define NO      8192
#define INV_T   0.125f
#define LN_EPS  1e-6f

#define WAVES   4
#define FBLOCK  (WAVES*32)
#define NB      128      // obs rows per LDS stage
#define LDK     68       // padded LDS stride for K rows (bank-conflict free A-frag reads)
#define LDV     40       // padded LDS stride for gated-V rows
#define LDA     33       // padded stride for agg transpose

#ifdef __has_builtin
# if __has_builtin(__builtin_amdgcn_global_load_async_to_lds_b128)
#  define HAVE_ASYNC_BUILTIN 1
# endif
# if __has_builtin(__builtin_amdgcn_s_wait_asynccnt)
#  define HAVE_WAITASYNC_BUILTIN 1
# endif
#endif

__device__ __forceinline__ float sigmoidf_(float x) { return 1.0f / (1.0f + __expf(-x)); }

// ---- async global->LDS copy of one float4 (ASYNCcnt-tracked) ----
__device__ __forceinline__ void async_cp16(void* lds_generic, const void* g) {
#if defined(HAVE_ASYNC_BUILTIN)
  __builtin_amdgcn_global_load_async_to_lds_b128(
      (as1_v4i_p)(unsigned long long)g,
      (as3_v4i_p)(unsigned)(unsigned long long)lds_generic,
      0, 0);
#else
  unsigned lds_off = (unsigned)(unsigned long long)lds_generic;
  asm volatile("global_load_async_to_lds_b128 %0, %1, off"
               :: "v"(lds_off), "v"(g) : "memory");
#endif
}
__device__ __forceinline__ void wait_async_le24() {
#if defined(HAVE_WAITASYNC_BUILTIN)
  __builtin_amdgcn_s_wait_asynccnt((short)24);
#else
  asm volatile("s_wait_asynccnt 0x18" ::: "memory");
#endif
}
__device__ __forceinline__ void wait_async_le0() {
#if defined(HAVE_WAITASYNC_BUILTIN)
  __builtin_amdgcn_s_wait_asynccnt((short)0);
#else
  asm volatile("s_wait_asynccnt 0x0" ::: "memory");
#endif
}

// issue the async copies for one obs block (24 x b128 per thread)
__device__ __forceinline__ void stage_block(int tid,
                                            const float* __restrict__ obsk,
                                            const float* __restrict__ gobs,
                                            int jb, float* lk_base, float* lv_base)
{
  // K: 128 rows x 64 f32 = 2048 float4; 16 threads per row -> coalesced
#pragma unroll
  for (int p = 0; p < 16; ++p) {
    int n   = p * FBLOCK + tid;
    int row = n >> 4, col = (n & 15) * 4;
    async_cp16(lk_base + row * LDK + col,
               obsk + (size_t)(jb + row) * EMBED + col);
  }
  // V: 128 rows x 32 f32 = 1024 float4; 8 threads per row
#pragma unroll
  for (int p = 0; p < 8; ++p) {
    int n   = p * FBLOCK + tid;
    int row = n >> 3, col = (n & 7) * 4;
    async_cp16(lv_base + row * LDV + col,
               gobs + (size_t)(jb + row) * 32 + col);
  }
}

// ---------------------------------------------------------------------------
// Kernel 1: projections. 16 lanes per row; lane c owns output cols 4c..4c+3.
// ---------------------------------------------------------------------------
template <bool IS_MAP>
__global__ __launch_bounds__(256) void proj_kernel(
    const float* __restrict__ x,
    const float* __restrict__ Wq, const float* __restrict__ Wk, const float* __restrict__ Wv,
    float* __restrict__ q_out, float* __restrict__ k_out,
    float* __restrict__ self_out, float* __restrict__ gated_out, int n_rows)
{
  int gid = blockIdx.x * blockDim.x + threadIdx.x;
  int row = gid >> 4;
  int c   = gid & 15;
  if (row >= n_rows) return;
  const float* xr = x + row * EMBED;

  float4 q = {0,0,0,0}, k = {0,0,0,0}, v = {0,0,0,0};
  for (int kk = 0; kk < EMBED; ++kk) {
    float xv = xr[kk];
    float4 wk = *(const float4*)(Wk + kk*EMBED + 4*c);
    float4 wv = *(const float4*)(Wv + kk*EMBED + 4*c);
    k.x += xv*wk.x; k.y += xv*wk.y; k.z += xv*wk.z; k.w += xv*wk.w;
    v.x += xv*wv.x; v.y += xv*wv.y; v.z += xv*wv.z; v.w += xv*wv.w;
    if (IS_MAP) {
      float4 wq = *(const float4*)(Wq + kk*EMBED + 4*c);
      q.x += xv*wq.x; q.y += xv*wq.y; q.z += xv*wq.z; q.w += xv*wq.w;
    }
  }

  if (IS_MAP) {
    *(float4*)(q_out + row*EMBED + 4*c) = q;
    float s = q.x*k.x + q.y*k.y + q.z*k.z + q.w*k.w;
    s += __shfl_xor(s, 1, 32); s += __shfl_xor(s, 2, 32);
    s += __shfl_xor(s, 4, 32); s += __shfl_xor(s, 8, 32);
    if (c == 0) self_out[row] = s;
  } else {
    *(float4*)(k_out + row*EMBED + 4*c) = k;
  }

  // GLU: gated[e] = v[e] * sigmoid(v[e+32]); partner lane is c^8
  float4 pv;
  pv.x = __shfl_xor(v.x, 8, 32); pv.y = __shfl_xor(v.y, 8, 32);
  pv.z = __shfl_xor(v.z, 8, 32); pv.w = __shfl_xor(v.w, 8, 32);
  if (c < 8) {
    float4 g;
    g.x = v.x * sigmoidf_(pv.x); g.y = v.y * sigmoidf_(pv.y);
    g.z = v.z * sigmoidf_(pv.z); g.w = v.w * sigmoidf_(pv.w);
    *(float4*)(gated_out + row*32 + 4*c) = g;
  }
}

// ---------------------------------------------------------------------------
// Kernel 2: fused flash attention + output proj + residual + layernorm.
// Double-buffered async LDS staging; fp32 WMMA for both GEMMs.
// ---------------------------------------------------------------------------
__global__ __launch_bounds__(FBLOCK) void flash_kernel(
    const float* __restrict__ mapq, const float* __restrict__ selfsc,
    const float* __restrict__ gmap, const float* __restrict__ obsk,
    const float* __restrict__ gobs, const float* __restrict__ map_code,
    const float* __restrict__ Wo,   const float* __restrict__ bo,
    const float* __restrict__ gamma,const float* __restrict__ beta,
    float* __restrict__ out)
{
  __shared__ float lds_k0[NB * LDK];
  __shared__ float lds_k1[NB * LDK];
  __shared__ float lds_v0[NB * LDV];
  __shared__ float lds_v1[NB * LDV];
  __shared__ float lds_agg[WAVES * 16 * LDA];
  __shared__ float lds_wo[32 * EMBED];
  __shared__ float lds_b[3 * EMBED];

  const int tid  = threadIdx.x;
  const int lane = tid & 31;
  const int wv   = tid >> 5;
  const int l16  = lane & 15;
  const int hi   = lane >> 4;                  // 0: lanes 0-15, 1: lanes 16-31
  const int row_base = blockIdx.x * (WAVES * 16) + wv * 16;

  for (int i = tid; i < 32 * EMBED; i += FBLOCK) lds_wo[i] = Wo[i];
  for (int i = tid; i < EMBED; i += FBLOCK) {
    lds_b[i] = bo[i]; lds_b[EMBED + i] = gamma[i]; lds_b[2*EMBED + i] = beta[i];
  }

  // prefetch first obs block into buffer 0
  stage_block(tid, obsk, gobs, 0, lds_k0, lds_v0);

  // Q as B-fragments (B[k][m]: lane -> col m=l16, vgpr pair -> k = 4c + 2*hi + {0,1})
  v2f bq[16];
  {
    const float* qr = mapq + (row_base + l16) * EMBED + 2 * hi;
#pragma unroll
    for (int c2 = 0; c2 < 16; ++c2) {
      float2 t = *(const float2*)(qr + 4 * c2);
      bq[c2].x = t.x * INV_T; bq[c2].y = t.y * INV_T;
    }
  }

  float mrun = selfsc[row_base + l16] * INV_T;  // self term seeds the softmax
  float lrun = 1.0f;
  v8f acc0, acc1;                               // agg' layout: (e = r + 8*hi [+16], m = l16)
  {
    const float* g = gmap + (row_base + l16) * 32;
#pragma unroll
    for (int r = 0; r < 8; ++r) {
      acc0[r] = g[r + 8*hi];
      acc1[r] = g[16 + r + 8*hi];
    }
  }

  for (int jb = 0; jb < NO; jb += NB) {
    const int  cur  = (jb / NB) & 1;
    const bool more = (jb + NB) < NO;
    float* lk_c = cur ? lds_k1 : lds_k0;
    float* lv_c = cur ? lds_v1 : lds_v0;

    __syncthreads();                         // prev-buffer readers all done
    if (more)
      stage_block(tid, obsk, gobs, jb + NB, cur ? lds_k0 : lds_k1,
                                            cur ? lds_v0 : lds_v1);
    if (more) wait_async_le24(); else wait_async_le0();   // current buffer landed
    __syncthreads();                         // visible to all waves

    for (int jt = 0; jt < NB; jt += 16) {
      // S' = K_tile(16x64) @ Q^T : C layout (j = r + 8*hi, m = l16)
      // two independent accumulation chains for ILP
      v8f sa = {0,0,0,0,0,0,0,0};
      v8f sb = {0,0,0,0,0,0,0,0};
      const float* lk = lk_c + (jt + l16) * LDK + 2 * hi;
#pragma unroll
      for (int c2 = 0; c2 < 8; ++c2) {
        float2 ta = *(const float2*)(lk + 4 * c2);
        float2 tb = *(const float2*)(lk + 4 * (c2 + 8));
        v2f a0; a0.x = ta.x; a0.y = ta.y;
        v2f a1; a1.x = tb.x; a1.y = tb.y;
        sa = __builtin_amdgcn_wmma_f32_16x16x4_f32(false, a0, false, bq[c2],
                                                   (short)0, sa, false, false);
        sb = __builtin_amdgcn_wmma_f32_16x16x4_f32(false, a1, false, bq[c2 + 8],
                                                   (short)0, sb, false, false);
      }
      v8f s = sa + sb;

      // online softmax over obs dim (column m = l16 lives per-lane, halves replicated)
      float cmax = s[0];
#pragma unroll
      for (int r = 1; r < 8; ++r) cmax = fmaxf(cmax, s[r]);
      cmax = fmaxf(cmax, __shfl_xor(cmax, 16, 32));
      float mnew  = fmaxf(mrun, cmax);
      float scale = __expf(mrun - mnew);
      float csum = 0.0f;
      v8f p;
#pragma unroll
      for (int r = 0; r < 8; ++r) { p[r] = __expf(s[r] - mnew); csum += p[r]; }
      csum += __shfl_xor(csum, 16, 32);
      lrun = lrun * scale + csum;
      mrun = mnew;
#pragma unroll
      for (int r = 0; r < 8; ++r) { acc0[r] *= scale; acc1[r] *= scale; }

      // agg' += V^T(16x4) x P'(4x16); P B-fragments rebuilt via xor16 shuffles
#pragma unroll
      for (int c2 = 0; c2 < 4; ++c2) {
        v2f b;
        if (c2 < 2) {
          float t2 = __shfl_xor(p[4*c2 + 2], 16, 32);
          float t3 = __shfl_xor(p[4*c2 + 3], 16, 32);
          b.x = hi ? t2 : p[4*c2];
          b.y = hi ? t3 : p[4*c2 + 1];
        } else {
          float t0 = __shfl_xor(p[4*c2 - 8], 16, 32);
          float t1 = __shfl_xor(p[4*c2 - 7], 16, 32);
          b.x = hi ? p[4*c2 - 6] : t0;
          b.y = hi ? p[4*c2 - 5] : t1;
        }
        const float* lv = lv_c + (jt + 4*c2 + 2*hi) * LDV;
        v2f a0; a0.x = lv[l16];      a0.y = lv[LDV + l16];        // e = 0..15
        acc0 = __builtin_amdgcn_wmma_f32_16x16x4_f32(false, a0, false, b,
                                                     (short)0, acc0, false, false);
        v2f a1; a1.x = lv[16 + l16]; a1.y = lv[LDV + 16 + l16];   // e = 16..31
        acc1 = __builtin_amdgcn_wmma_f32_16x16x4_f32(false, a1, false, b,
                                                     (short)0, acc1, false, false);
      }
    }
  }

  // ---- epilogue: normalize, transpose agg through LDS, out-proj + LN ----
  float inv = 1.0f / lrun;
  {
    float* la = lds_agg + wv * 16 * LDA + l16 * LDA;
#pragma unroll
    for (int r = 0; r < 8; ++r) {
      la[r + 8*hi]      = acc0[r] * inv;
      la[16 + r + 8*hi] = acc1[r] * inv;
    }
  }
  __syncthreads();
  {
    int m = lane >> 1;                  // 2 lanes per map row
    int ebase = (lane & 1) * 32;
    int row_g = row_base + m;
    const float* ar = lds_agg + wv * 16 * LDA + m * LDA;
    float ag[32];
#pragma unroll
    for (int h = 0; h < 32; ++h) ag[h] = ar[h];
    float o[32];
#pragma unroll
    for (int e = 0; e < 32; ++e) {
      int ee = ebase + e;
      float acc = lds_b[ee] + map_code[(size_t)row_g * EMBED + ee];
      for (int h = 0; h < 32; ++h) acc += ag[h] * lds_wo[h * EMBED + ee];
      o[e] = acc;
    }
    float s1 = 0.0f;
#pragma unroll
    for (int e = 0; e < 32; ++e) s1 += o[e];
    s1 += __shfl_xor(s1, 1, 32);
    float mu = s1 * (1.0f / 64.0f);
    float s2 = 0.0f;
#pragma unroll
    for (int e = 0; e < 32; ++e) { float d = o[e] - mu; s2 += d * d; }
    s2 += __shfl_xor(s2, 1, 32);
    float rstd = rsqrtf(s2 * (1.0f / 64.0f) + LN_EPS);
#pragma unroll
    for (int e = 0; e < 32; ++e) {
      int ee = ebase + e;
      out[(size_t)row_g * EMBED + ee] =
          (o[e] - mu) * rstd * lds_b[EMBED + ee] + lds_b[2*EMBED + ee];
    }
  }
}

// ---------------------------------------------------------------------------
extern "C" void kernel_launch(void* const* d_in, const int* in_sizes, int n_in,
                              void* d_out, int out_size, void* d_ws, size_t ws_size,
                              hipStream_t stream)
{
  const float* map_code = (const float*)d_in[0];
  const float* obs_code = (const float*)d_in[1];
  const float* Wq    = (const float*)d_in[2];
  const float* Wk    = (const float*)d_in[3];
  const float* Wv    = (const float*)d_in[4];
  const float* Wo    = (const float*)d_in[5];
  const float* bo    = (const float*)d_in[6];
  const float* gamma = (const float*)d_in[7];
  const float* beta  = (const float*)d_in[8];
  float* out = (float*)d_out;

  float* ws     = (float*)d_ws;
  float* mapq   = ws;                            // NM*64
  float* selfsc = mapq   + (size_t)NM * EMBED;   // NM
  float* gmap   = selfsc + NM;                   // NM*32
  float* obsk   = gmap   + (size_t)NM * 32;      // NO*64
  float* gobs   = obsk   + (size_t)NO * EMBED;   // NO*32

  proj_kernel<true ><<<NM * 16 / 256, 256, 0, stream>>>(
      map_code, Wq, Wk, Wv, mapq, nullptr, selfsc, gmap, NM);
  proj_kernel<false><<<NO * 16 / 256, 256, 0, stream>>>(
      obs_code, Wq, Wk, Wv, nullptr, obsk, nullptr, gobs, NO);

  flash_kernel<<<NM / (WAVES * 16), FBLOCK, 0, stream>>>(
      mapq, selfsc, gmap, obsk, gobs, map_code, Wo, bo, gamma, beta, out);
}